// ETimesNet_23295902614287
// MI455X (gfx1250) — compile-verified
//
#include <hip/hip_runtime.h>
#include <hip/hip_bf16.h>

#define TT     64
#define TT1    65
#define NGRP   128
#define NNODE  10
#define NTOT   1280       // NGRP*NNODE
#define HID    16
#define KTOP   2
#define EDGES  11520      // NGRP*NNODE*(NNODE-1)
#define HCAP   128        // max padded h length (T1=65 -> worst pad = 2*64 = 128)
#define HSTR   (HCAP*HID) // 2048 floats per node in the h buffers
#define WPB    8          // waves per block for WMMA kernels

typedef float v8f __attribute__((ext_vector_type(8)));
typedef float v2f __attribute__((ext_vector_type(2)));

// silu via v_rcp_f32 instead of the IEEE-divide expansion (big VALU saving)
__device__ __forceinline__ float silu_f(float v) {
    return v * __builtin_amdgcn_rcpf(1.0f + __expf(-v));
}

// D = A(16x4 f32) * B(4x16 f32) + C, full-wave fp32 WMMA
__device__ __forceinline__ v8f wmma4(v2f a, v2f b, v8f c) {
    return __builtin_amdgcn_wmma_f32_16x16x4_f32(false, a, false, b, (short)0, c, false, false);
}

// A-fragment chunk c from an LDS tile [16][stride]:
// lanes 0-15: row=lane, K={4c,4c+1}; lanes 16-31: row=lane-16, K={4c+2,4c+3}
__device__ __forceinline__ v2f afrag_lds(const float* buf, int stride, int c, int lane) {
    const int r = lane & 15;
    const int k = 4 * c + ((lane >> 4) << 1);
    v2f a; a.x = buf[r * stride + k]; a.y = buf[r * stride + k + 1]; return a;
}
// B-fragment chunk c for Y = X * W^T with W row-major [16 out][stride in]:
// B[k][n] = W[n][k]; same K striping as A.
__device__ __forceinline__ v2f bfrag_w(const float* __restrict__ W, int stride, int c,
                                       int lane, float s) {
    const int n = lane & 15;
    const int k = 4 * c + ((lane >> 4) << 1);
    v2f b; b.x = s * W[n * stride + k]; b.y = s * W[n * stride + k + 1]; return b;
}

// ---------------------------------------------------------------------------
// init: TEP[t1,d] = sum_t time_emb[t,d]*predW[t1,t] + predb[t1]; SROW[t1]=sum_t predW
__global__ void init_te_kernel(const float* __restrict__ temb, const float* __restrict__ predW,
                               const float* __restrict__ predb, float* __restrict__ tep,
                               float* __restrict__ srow) {
    const int idx = blockIdx.x * 32 + threadIdx.x;
    if (idx >= TT1 * HID) return;
    const int t1 = idx / HID, d = idx % HID;
    float a = 0.f;
    for (int t = 0; t < TT; t++) a += temb[t * HID + d] * predW[t1 * TT + t];
    tep[idx] = a + predb[t1];
    if (d == 0) {
        float s = 0.f;
        for (int t = 0; t < TT; t++) s += predW[t1 * TT + t];
        srow[t1] = s;
    }
}

// H[n,t1,d] = (h0[n]*embW[d]+embb[d])*SROW[t1] + TEP[t1,d]   (pred-projected h)
__global__ void init_h_kernel(const float* __restrict__ h0, const float* __restrict__ embW,
                              const float* __restrict__ embb, const float* __restrict__ tep,
                              const float* __restrict__ srow, float* __restrict__ H) {
    const int idx = blockIdx.x * 256 + threadIdx.x; // N*T1*HID exactly
    const int d = idx & 15;
    const int q = idx >> 4;
    const int t = q % TT1;
    const int n = q / TT1;
    const float A = h0[n] * embW[d] + embb[d];
    H[n * HSTR + t * HID + d] = A * srow[t] + tep[t * HID + d];
}

// X[n,t,c] node-major with x[:,T]=x[:,T-1] appended
__global__ void init_x_kernel(const float* __restrict__ x0, float* __restrict__ X) {
    const int idx = blockIdx.x * 256 + threadIdx.x; // N*T1*3 exactly
    const int c = idx % 3;
    const int q = idx / 3;
    const int t = q % TT1;
    const int n = q / TT1;
    const int ts = (t == TT) ? (TT - 1) : t;
    X[idx] = x0[(ts * NTOT + n) * 3 + c];
}

// pw[n,k]: softmax_k of sqrt(mean_c |DFT_f(x_centered)|^2) at f = top_list[n%NN, k]
__global__ void pw_kernel(const float* __restrict__ x0, const int* __restrict__ tl,
                          float* __restrict__ pw) {
    const int n = blockIdx.x * 32 + threadIdx.x;
    if (n >= NTOT) return;
    const int g = n / NNODE, nn = n % NNODE;
    float vals[KTOP];
    for (int k = 0; k < KTOP; k++) {
        const int f = tl[nn * KTOP + k];
        float acc = 0.f;
        for (int c = 0; c < 3; c++) {
            float re = 0.f, im = 0.f;
            for (int t = 0; t < TT; t++) {
                float mean = 0.f;
                for (int j = 0; j < NNODE; j++) mean += x0[(t * NTOT + g * NNODE + j) * 3 + c];
                mean *= (1.0f / (float)NNODE);
                const float v = x0[(t * NTOT + n) * 3 + c] - mean;
                const float ang = -6.283185307179586f * (float)(f * t) / (float)TT;
                re += v * __cosf(ang);
                im += v * __sinf(ang);
            }
            acc += re * re + im * im;
        }
        vals[k] = sqrtf(acc * (1.0f / 3.0f));
    }
    const float mx = fmaxf(vals[0], vals[1]);
    const float e0 = __expf(vals[0] - mx), e1 = __expf(vals[1] - mx);
    const float inv = __builtin_amdgcn_rcpf(e0 + e1);
    pw[n * KTOP + 0] = e0 * inv;
    pw[n * KTOP + 1] = e1 * inv;
}

// ---------------------------------------------------------------------------
// EGCL edge phase: one wave per (node,t); its 9 outgoing edges are tile rows 0..8.
// Chain: e_in(36) -> eW1 -> silu -> eW2 -> silu = m ; m -> cW1 -> silu -> cW2 -> scale.
// magg/agg reduced in-wave (deterministic, no atomics). Writes x_out and magg.
__global__ __launch_bounds__(32 * WPB) void egcl_edge_kernel(
    const float* __restrict__ h_in, const float* __restrict__ x_in,
    const int* __restrict__ edges, const float* __restrict__ eattr,
    const float* __restrict__ eW1, const float* __restrict__ eb1,
    const float* __restrict__ eW2, const float* __restrict__ eb2,
    const float* __restrict__ cW1, const float* __restrict__ cb1,
    const float* __restrict__ cW2,
    float* __restrict__ x_out, float* __restrict__ magg) {
    __shared__ float S[WPB * 16 * 40];
    __shared__ float M[WPB * 16 * 17];
    __shared__ float SC[WPB * 16];
    __shared__ float TB[WPB * 16 * 4];
    const int lane = threadIdx.x & 31;
    const int w = threadIdx.x >> 5;
    float* Sw = S + w * (16 * 40);
    float* Mw = M + w * (16 * 17);
    float* SCw = SC + w * 16;
    float* TBw = TB + w * 64;
    const int wid = blockIdx.x * WPB + w;   // 0 .. N*T1-1 (exact grid)
    const int n = wid / TT1;
    const int t = wid % TT1;
    const int col = lane & 15;

    float dnx = 0.f, dny = 0.f, dnz = 0.f;
    if (lane < 16) {
        const int r = lane;
        if (r < NNODE - 1) {
            const int e = n * (NNODE - 1) + r;
            const int cn = edges[EDGES + e];  // col (dst) node
            const float xr0 = x_in[(n * TT1 + t) * 3 + 0];
            const float xr1 = x_in[(n * TT1 + t) * 3 + 1];
            const float xr2 = x_in[(n * TT1 + t) * 3 + 2];
            const float xc0 = x_in[(cn * TT1 + t) * 3 + 0];
            const float xc1 = x_in[(cn * TT1 + t) * 3 + 1];
            const float xc2 = x_in[(cn * TT1 + t) * 3 + 2];
            const float dx = xr0 - xc0, dy = xr1 - xc1, dz = xr2 - xc2;
            const float radial = dx * dx + dy * dy + dz * dz;
            const float inv = __builtin_amdgcn_rcpf(sqrtf(radial) + 1.0f);
            dnx = dx * inv; dny = dy * inv; dnz = dz * inv;
            const float* hr = h_in + n * HSTR + t * HID;
            #pragma unroll
            for (int j = 0; j < 16; j++) Sw[r * 40 + j] = hr[j];
            Sw[r * 40 + 32] = radial;
            Sw[r * 40 + 33] = eattr[e * 3 + 0];
            Sw[r * 40 + 34] = eattr[e * 3 + 1];
            Sw[r * 40 + 35] = eattr[e * 3 + 2];
        } else {
            #pragma unroll
            for (int j = 0; j < 16; j++) Sw[r * 40 + j] = 0.f;
            Sw[r * 40 + 32] = 0.f; Sw[r * 40 + 33] = 0.f;
            Sw[r * 40 + 34] = 0.f; Sw[r * 40 + 35] = 0.f;
        }
    } else {
        const int r = lane - 16;
        if (r < NNODE - 1) {
            const int e = n * (NNODE - 1) + r;
            const int cn = edges[EDGES + e];
            const float* hc = h_in + cn * HSTR + t * HID;
            #pragma unroll
            for (int j = 0; j < 16; j++) Sw[r * 40 + 16 + j] = hc[j];
        } else {
            #pragma unroll
            for (int j = 0; j < 16; j++) Sw[r * 40 + 16 + j] = 0.f;
        }
    }
    __syncthreads();

    v8f a1 = {};
    #pragma unroll
    for (int c = 0; c < 9; c++)
        a1 = wmma4(afrag_lds(Sw, 40, c, lane), bfrag_w(eW1, 36, c, lane, 1.f), a1);
    {
        const float bv = eb1[col];
        const int mo = (lane >> 4) * 8;
        #pragma unroll
        for (int j = 0; j < 8; j++) Mw[(mo + j) * 17 + col] = silu_f(a1[j] + bv);
    }
    __syncthreads();

    v8f a2 = {};
    #pragma unroll
    for (int c = 0; c < 4; c++)
        a2 = wmma4(afrag_lds(Mw, 17, c, lane), bfrag_w(eW2, 16, c, lane, 1.f), a2);
    v8f mv;
    {
        const float bv = eb2[col];
        #pragma unroll
        for (int j = 0; j < 8; j++) mv[j] = silu_f(a2[j] + bv);
    }
    // magg[n,t,col] = sum over the 9 valid edge rows (rows 0..7 local + row 8 from upper half)
    {
        const float part = mv[0] + mv[1] + mv[2] + mv[3] + mv[4] + mv[5] + mv[6] + mv[7];
        const float row8 = __shfl(mv[0], col + 16, 32);
        if (lane < 16) magg[(n * TT1 + t) * HID + col] = part + row8;
    }
    __syncthreads();
    {
        const int mo = (lane >> 4) * 8;
        #pragma unroll
        for (int j = 0; j < 8; j++) Mw[(mo + j) * 17 + col] = mv[j];
    }
    __syncthreads();

    v8f a3 = {};
    #pragma unroll
    for (int c = 0; c < 4; c++)
        a3 = wmma4(afrag_lds(Mw, 17, c, lane), bfrag_w(cW1, 16, c, lane, 1.f), a3);
    float p[8];
    {
        const float bv = cb1[col];
        const float wv = cW2[col];
        #pragma unroll
        for (int j = 0; j < 8; j++) p[j] = silu_f(a3[j] + bv) * wv;
    }
    #pragma unroll
    for (int off = 1; off < 16; off <<= 1) {
        #pragma unroll
        for (int j = 0; j < 8; j++) p[j] += __shfl_xor(p[j], off, 32);
    }
    if (lane == 0) {
        #pragma unroll
        for (int j = 0; j < 8; j++) SCw[j] = p[j];
    }
    if (lane == 16) {
        #pragma unroll
        for (int j = 0; j < 8; j++) SCw[8 + j] = p[j];
    }
    __syncthreads();
    if (lane < NNODE - 1) {
        const float sc = SCw[lane];
        TBw[lane * 4 + 0] = fminf(fmaxf(dnx * sc, -100.f), 100.f);
        TBw[lane * 4 + 1] = fminf(fmaxf(dny * sc, -100.f), 100.f);
        TBw[lane * 4 + 2] = fminf(fmaxf(dnz * sc, -100.f), 100.f);
    }
    __syncthreads();
    if (lane < 3) {
        float a = 0.f;
        #pragma unroll
        for (int r = 0; r < NNODE - 1; r++) a += TBw[r * 4 + lane];
        const int xi = (n * TT1 + t) * 3 + lane;
        x_out[xi] = x_in[xi] + a * (1.0f / (float)(NNODE - 1)); // cnt = 9 per node
    }
}

// EGCL node phase: h_out = h + nW2( silu( nW1([h, magg]) ) ); full 16-row tiles.
__global__ __launch_bounds__(32 * WPB) void egcl_node_kernel(
    const float* __restrict__ h_in, const float* __restrict__ magg,
    const float* __restrict__ nW1, const float* __restrict__ nb1,
    const float* __restrict__ nW2, const float* __restrict__ nb2,
    float* __restrict__ h_out) {
    __shared__ float S[WPB * 16 * 36];
    __shared__ float M[WPB * 16 * 17];
    const int lane = threadIdx.x & 31;
    const int w = threadIdx.x >> 5;
    float* Sw = S + w * (16 * 36);
    float* Mw = M + w * (16 * 17);
    const int q0 = (blockIdx.x * WPB + w) * 16;   // exact grid over N*T1
    const int col = lane & 15;
    if (lane < 16) {
        const int q = q0 + lane;
        const int n = q / TT1, t = q % TT1;
        const float* hp = h_in + n * HSTR + t * HID;
        #pragma unroll
        for (int j = 0; j < 16; j++) Sw[lane * 36 + j] = hp[j];
    } else {
        const int r = lane - 16;
        const float* mp = magg + (q0 + r) * HID;
        #pragma unroll
        for (int j = 0; j < 16; j++) Sw[r * 36 + 16 + j] = mp[j];
    }
    __syncthreads();
    v8f a1 = {};
    #pragma unroll
    for (int c = 0; c < 8; c++)
        a1 = wmma4(afrag_lds(Sw, 36, c, lane), bfrag_w(nW1, 32, c, lane, 1.f), a1);
    {
        const float bv = nb1[col];
        const int mo = (lane >> 4) * 8;
        #pragma unroll
        for (int j = 0; j < 8; j++) Mw[(mo + j) * 17 + col] = silu_f(a1[j] + bv);
    }
    __syncthreads();
    v8f a2 = {};
    #pragma unroll
    for (int c = 0; c < 4; c++)
        a2 = wmma4(afrag_lds(Mw, 17, c, lane), bfrag_w(nW2, 16, c, lane, 1.f), a2);
    {
        const float bv = nb2[col];
        const int mo = (lane >> 4) * 8;
        #pragma unroll
        for (int j = 0; j < 8; j++) {
            const int q = q0 + mo + j;
            const int n = q / TT1, t = q % TT1;
            const int idx = n * HSTR + t * HID + col;
            h_out[idx] = h_in[idx] + a2[j] + bv;
        }
    }
}

// ---------------------------------------------------------------------------
// TimesNet control: ctrl[0]=len, [1..10]=pr[nn], [11..20]=s1[nn], [21]=max_len, [22]=oldLen
__global__ void tn_prep_kernel(const int* __restrict__ tl, int* __restrict__ ctrl, int k) {
    if (threadIdx.x == 0 && blockIdx.x == 0) {
        const int oldLen = (k == 0) ? TT1 : ctrl[0];
        int maxlen = 0;
        int prv[NNODE];
        for (int nn = 0; nn < NNODE; nn++) {
            int tv = tl[nn * KTOP + k];
            if (tv < 1) tv = 1;
            int pr = TT / tv;
            if (pr < 1) pr = 1;
            prv[nn] = pr;
            if ((TT1 % pr) != 0) {
                const int ml = (TT1 / pr + 1) * pr;
                if (ml > maxlen) maxlen = ml;
            }
        }
        const int newLen = (maxlen > oldLen) ? maxlen : oldLen;
        ctrl[0] = newLen;
        ctrl[21] = maxlen;
        ctrl[22] = oldLen;
        for (int nn = 0; nn < NNODE; nn++) {
            ctrl[1 + nn] = prv[nn];
            ctrl[11 + nn] = (maxlen / prv[nn]) * prv[nn];
        }
    }
}

// Pad h rows [oldLen, newLen) with row oldLen-1 (worst-case fixed grid, ctrl-gated)
__global__ void tn_pad_kernel(float* __restrict__ h, const int* __restrict__ ctrl) {
    const int idx = blockIdx.x * 256 + threadIdx.x;  // N*HCAP*HID exactly
    const int d = idx & 15;
    const int t = (idx >> 4) & (HCAP - 1);
    const int n = idx >> 11;
    const int oldL = ctrl[22], newL = ctrl[0];
    if (t >= oldL && t < newL)
        h[n * HSTR + t * HID + d] = h[n * HSTR + (oldL - 1) * HID + d];
}

// Fused inception: out = 0.5*conv1x1 + 0.5*conv3x3(SAME) + 0.5*(b1+b2) on the (r,pr)
// grid of the first s1 timesteps; passthrough elsewhere. 40-chunk f32 WMMA chain.
__global__ __launch_bounds__(32 * WPB) void tn_conv_kernel(
    const float* __restrict__ h_in, float* __restrict__ h_out,
    float* __restrict__ res, const int* __restrict__ ctrl,
    const float* __restrict__ cw1, const float* __restrict__ cb1,
    const float* __restrict__ cw2, const float* __restrict__ cb2) {
    const int lane = threadIdx.x & 31;
    const int w = threadIdx.x >> 5;
    const int wid = blockIdx.x * WPB + w;   // N * (HCAP/16) waves
    const int n = wid >> 3;
    const int t0 = (wid & 7) * 16;
    const int len = ctrl[0];
    if (t0 >= len) return;                  // no barriers in this kernel
    const int nn = n % NNODE;
    const int pr = ctrl[1 + nn];
    const int s1 = ctrl[11 + nn];
    const int rr = s1 / pr;
    const int col = lane & 15;
    const int kadd = (lane >> 4) << 1;
    const int tA = t0 + col;                // A-row owned by this lane
    const int yy = tA / pr;
    const int xx = tA % pr;
    const bool base_ok = (tA < s1);
    const float* hbase = h_in + n * HSTR;

    v8f acc = {};
    #pragma unroll
    for (int c = 0; c < 4; c++) {           // 1x1 conv taps
        const int ic = 4 * c + kadd;
        v2f a;
        if (base_ok) { a.x = hbase[tA * HID + ic]; a.y = hbase[tA * HID + ic + 1]; }
        else         { a.x = 0.f; a.y = 0.f; }
        v2f b;
        b.x = 0.5f * cw1[col * HID + ic];
        b.y = 0.5f * cw1[col * HID + ic + 1];
        acc = wmma4(a, b, acc);
    }
    #pragma unroll
    for (int ky = 0; ky < 3; ky++) {
        #pragma unroll
        for (int kx = 0; kx < 3; kx++) {
            const int y2 = yy + ky - 1;
            const int x2 = xx + kx - 1;
            const bool ok = base_ok && (y2 >= 0) && (y2 < rr) && (x2 >= 0) && (x2 < pr);
            const int tp = y2 * pr + x2;
            #pragma unroll
            for (int c = 0; c < 4; c++) {
                const int ic = 4 * c + kadd;
                v2f a;
                if (ok) { a.x = hbase[tp * HID + ic]; a.y = hbase[tp * HID + ic + 1]; }
                else    { a.x = 0.f; a.y = 0.f; }
                v2f b;
                b.x = 0.5f * cw2[((col * HID + ic) * 3 + ky) * 3 + kx];
                b.y = 0.5f * cw2[((col * HID + ic + 1) * 3 + ky) * 3 + kx];
                acc = wmma4(a, b, acc);
            }
        }
    }
    const float bias = 0.5f * (cb1[col] + cb2[col]);
    const int mo = (lane >> 4) * 8;
    #pragma unroll
    for (int j = 0; j < 8; j++) {
        const int t = t0 + mo + j;
        if (t >= len) continue;
        const float hv = hbase[t * HID + col];
        const float val = (t < s1) ? (acc[j] + bias) : hv;
        h_out[n * HSTR + t * HID + col] = val;
        if (t < TT1) res[(n * TT1 + t) * HID + col] = val;
    }
}

// h_mix = pw0*res0 + pw1*res1 + h_conv[:, :T1]
__global__ void mix_kernel(const float* __restrict__ hA, const float* __restrict__ res,
                           const float* __restrict__ pw, float* __restrict__ hB) {
    const int idx = blockIdx.x * 256 + threadIdx.x;  // N*T1*HID exactly
    const int d = idx & 15;
    const int q = idx >> 4;
    const int t = q % TT1;
    const int n = q / TT1;
    const float r0 = res[idx];
    const float r1 = res[(size_t)NTOT * TT1 * HID + idx];
    hB[n * HSTR + t * HID + d] = pw[n * KTOP] * r0 + pw[n * KTOP + 1] * r1
                               + hA[n * HSTR + t * HID + d];
}

// out[n,c] = sum_t theta[t]*(x[n,t,c]-x[n,63,c]) + x[n,63,c]
__global__ void final_kernel(const float* __restrict__ X, const float* __restrict__ theta,
                             float* __restrict__ out) {
    const int idx = blockIdx.x * 32 + threadIdx.x;
    if (idx >= NTOT * 3) return;
    const int c = idx % 3, n = idx / 3;
    const float xl = X[(n * TT1 + (TT - 1)) * 3 + c];
    float a = 0.f;
    for (int t = 0; t < TT; t++) a += theta[t] * (X[(n * TT1 + t) * 3 + c] - xl);
    out[idx] = a + xl;
}

// ---------------------------------------------------------------------------
static void launch_egcl(const float* hin, float* hout, const float* xin, float* xout,
                        float* magg, const int* edges, const float* eattr,
                        const float* const* P, int l, bool do_node, hipStream_t stream) {
    egcl_edge_kernel<<<(NTOT * TT1) / WPB, 32 * WPB, 0, stream>>>(
        hin, xin, edges, eattr,
        P[0] + l * HID * 36, P[1] + l * HID, P[2] + l * HID * HID, P[3] + l * HID,
        P[8] + l * HID * HID, P[9] + l * HID, P[10] + l * HID,
        xout, magg);
    if (do_node)
        egcl_node_kernel<<<(NTOT * TT1) / (16 * WPB), 32 * WPB, 0, stream>>>(
            hin, magg,
            P[4] + l * HID * 2 * HID, P[5] + l * HID, P[6] + l * HID * HID, P[7] + l * HID,
            hout);
}

extern "C" void kernel_launch(void* const* d_in, const int* in_sizes, int n_in,
                              void* d_out, int out_size, void* d_ws, size_t ws_size,
                              hipStream_t stream) {
    (void)in_sizes; (void)n_in; (void)out_size;
    const float* h0    = (const float*)d_in[0];
    const float* x0    = (const float*)d_in[1];
    const int*   edges = (const int*)d_in[2];
    const float* eattr = (const float*)d_in[3];
    const int*   tl    = (const int*)d_in[4];
    const float* embW  = (const float*)d_in[5];
    const float* embb  = (const float*)d_in[6];
    const float* temb  = (const float*)d_in[7];
    const float* predW = (const float*)d_in[8];
    const float* predb = (const float*)d_in[9];
    const float* theta = (const float*)d_in[10];
    const float* cw1   = (const float*)d_in[11];
    const float* cb1   = (const float*)d_in[12];
    const float* cw2   = (const float*)d_in[13];
    const float* cb2   = (const float*)d_in[14];
    const float* pp[11];
    const float* mp[11];
    for (int i = 0; i < 11; i++) pp[i] = (const float*)d_in[15 + i];
    for (int i = 0; i < 11; i++) mp[i] = (const float*)d_in[26 + i];

    float* ws = (float*)d_ws;
    float* HA   = ws;
    float* HB   = HA + (size_t)NTOT * HSTR;
    float* XA   = HB + (size_t)NTOT * HSTR;
    float* XB   = XA + (size_t)NTOT * TT1 * 3;
    float* MAGG = XB + (size_t)NTOT * TT1 * 3;
    float* RES0 = MAGG + (size_t)NTOT * TT1 * HID;
    float* RES1 = RES0 + (size_t)NTOT * TT1 * HID;
    float* PW   = RES1 + (size_t)NTOT * TT1 * HID;
    float* TEP  = PW + (size_t)NTOT * KTOP;
    float* SROW = TEP + (size_t)TT1 * HID;
    int*   CTRL = (int*)(SROW + TT1);
    const size_t needed = (size_t)((CTRL + 32) - (int*)ws) * sizeof(int);
    if (ws_size < needed) return;

    // ---- embed + time-projection of h; x staging; spectral weights
    init_te_kernel<<<(TT1 * HID + 31) / 32, 32, 0, stream>>>(temb, predW, predb, TEP, SROW);
    init_h_kernel<<<(NTOT * TT1 * HID) / 256, 256, 0, stream>>>(h0, embW, embb, TEP, SROW, HA);
    init_x_kernel<<<(NTOT * TT1 * 3) / 256, 256, 0, stream>>>(x0, XA);
    pw_kernel<<<(NTOT + 31) / 32, 32, 0, stream>>>(x0, tl, PW);

    // ---- prior EGCL x2
    launch_egcl(HA, HB, XA, XB, MAGG, edges, eattr, pp, 0, true, stream);
    launch_egcl(HB, HA, XB, XA, MAGG, edges, eattr, pp, 1, true, stream);

    // ---- TimesNet middle (device-resolved data-dependent shapes)
    tn_prep_kernel<<<1, 32, 0, stream>>>(tl, CTRL, 0);
    tn_pad_kernel<<<(NTOT * HCAP * HID) / 256, 256, 0, stream>>>(HA, CTRL);
    tn_conv_kernel<<<(NTOT * (HCAP / 16)) / WPB, 32 * WPB, 0, stream>>>(
        HA, HB, RES0, CTRL, cw1, cb1, cw2, cb2);
    tn_prep_kernel<<<1, 32, 0, stream>>>(tl, CTRL, 1);
    tn_pad_kernel<<<(NTOT * HCAP * HID) / 256, 256, 0, stream>>>(HB, CTRL);
    tn_conv_kernel<<<(NTOT * (HCAP / 16)) / WPB, 32 * WPB, 0, stream>>>(
        HB, HA, RES1, CTRL, cw1, cb1, cw2, cb2);
    mix_kernel<<<(NTOT * TT1 * HID) / 256, 256, 0, stream>>>(HA, RES0, PW, HB);

    // ---- main EGCL x2 (last node update is dead: only x feeds the readout)
    launch_egcl(HB, HA, XA, XB, MAGG, edges, eattr, mp, 0, true, stream);
    launch_egcl(HA, HB, XB, XA, MAGG, edges, eattr, mp, 1, false, stream);

    // ---- theta readout
    final_kernel<<<(NTOT * 3 + 31) / 32, 32, 0, stream>>>(XA, theta, (float*)d_out);
}